// GatNet_64991445123385
// MI455X (gfx1250) — compile-verified
//
#include <hip/hip_runtime.h>
#include <math.h>

#define HEADS 8

typedef float v2f __attribute__((ext_vector_type(2)));
typedef float v8f __attribute__((ext_vector_type(8)));

// ---------------------------------------------------------------------------
// fp32 WMMA GEMM:  O[N, COUT] = X[N, CIN] @ W[CIN, COUT]
// CIN/COUT compile-time -> fully unrolled branch-free K loop of
// V_WMMA_F32_16X16X4_F32, loads/stores with immediate offsets.
// One 16x16 tile per wave, 2 waves per block. All lanes always reach every
// WMMA (EXEC all-ones requirement).
// ---------------------------------------------------------------------------
template <int CIN, int COUT>
__global__ void __launch_bounds__(64) gemm16x4_wmma(
    const float* __restrict__ X, const float* __restrict__ W,
    float* __restrict__ O, int N) {
  const int wave = threadIdx.x >> 5;
  const int lane = threadIdx.x & 31;
  const int m0 = (blockIdx.x * 2 + wave) * 16;
  const int n0 = blockIdx.y * 16;
  const int r  = lane & 15;
  const int kk = (lane >> 4) * 2;     // lanes 0-15 -> K{0,1}; 16-31 -> K{2,3}
  const int row  = m0 + r;            // A-fragment row
  const int rowc = row < N ? row : N - 1;  // branch-free clamp for loads
  const int col  = n0 + r;            // B-fragment col (COUT multiple of 16)
  const float* __restrict__ Xr = X + (size_t)rowc * CIN + kk;  // imm offs: k*4
  const float* __restrict__ Wc = W + (size_t)kk * COUT + col;  // imm offs: k*COUT*4

  v8f acc = {};
  constexpr int KFULL = (CIN / 4) * 4;

  #pragma unroll
  for (int k = 0; k < KFULL; k += 4) {
    v2f a, b;
    const float2 t = *(const float2*)(Xr + k);
    a.x = t.x; a.y = t.y;
    b.x = Wc[(size_t)k * COUT];
    b.y = Wc[(size_t)(k + 1) * COUT];
    acc = __builtin_amdgcn_wmma_f32_16x16x4_f32(
        false, a, false, b, (short)0, acc, false, false);
  }

  if (CIN & 3) {                      // compile-time; only CIN=18 instantiates
    const int ka = KFULL + kk;
    const bool ok0 = (ka < CIN);
    const bool ok1 = (ka + 1 < CIN);
    const int kc = ok0 ? KFULL : -kk;  // clamped so Xr+kc, Wc+kc*COUT in range
    v2f a, b;
    const float2 t = *(const float2*)(Xr + kc);
    a.x = ok0 ? t.x : 0.f;
    a.y = ok1 ? t.y : 0.f;
    const float w0 = Wc[(size_t)kc * COUT];
    const float w1 = Wc[(size_t)(ok1 ? kc + 1 : kc) * COUT];
    b.x = ok0 ? w0 : 0.f;
    b.y = ok1 ? w1 : 0.f;
    acc = __builtin_amdgcn_wmma_f32_16x16x4_f32(
        false, a, false, b, (short)0, acc, false, false);
  }

  // C/D layout: VGPR v -> row m0 + v (+8 for upper half-wave), col = n0+r
  const int mh = (lane >> 4) * 8;
  float* __restrict__ Op = O + (size_t)(m0 + mh) * COUT + col;
  if (m0 + 15 < N) {                  // wave-uniform fast path (always, N%16==0)
    #pragma unroll
    for (int v = 0; v < 8; ++v) Op[(size_t)v * COUT] = acc[v];
  } else {
    #pragma unroll
    for (int v = 0; v < 8; ++v)
      if (m0 + mh + v < N) Op[(size_t)v * COUT] = acc[v];
  }
}

// ---------------------------------------------------------------------------
// per (node, head): al_s = <h[n,h,:], a_src[h,:]>, al_d likewise
// ---------------------------------------------------------------------------
template <int C>
__global__ void attn_logits_k(const float* __restrict__ Hf,
                              const float* __restrict__ aS,
                              const float* __restrict__ aD,
                              float* __restrict__ als, float* __restrict__ ald,
                              int N) {
  int i = blockIdx.x * blockDim.x + threadIdx.x;
  if (i >= N * HEADS) return;
  int h = i & 7;
  int n = i >> 3;
  const float* hp = Hf + ((size_t)n * HEADS + h) * C;
  float s = 0.f, d = 0.f;
  #pragma unroll
  for (int c = 0; c < C; ++c) {
    float v = hp[c];
    s = fmaf(v, aS[h * C + c], s);
    d = fmaf(v, aD[h * C + c], d);
  }
  als[i] = s;
  ald[i] = d;
}

__global__ void fillk(float* __restrict__ p, float v, size_t n) {
  size_t i = (size_t)blockIdx.x * blockDim.x + threadIdx.x;
  if (i < n) p[i] = v;
}

// branch-free float atomic-max: single native integer atomic, no CAS loop
__device__ __forceinline__ void atomicMaxF(float* addr, float v) {
  int iv = __float_as_int(v);
  if (iv >= 0) atomicMax((int*)addr, iv);
  else         atomicMin((unsigned int*)addr, (unsigned int)iv);
}

__device__ __forceinline__ float lrelu02(float t) {
  return t > 0.f ? t : 0.2f * t;
}

__device__ __forceinline__ void edge_ends(const int* __restrict__ ei, int E,
                                          int e, int& s, int& d) {
  if (e < E) { s = ei[e]; d = ei[E + e]; }
  else       { s = e - E; d = s; }          // appended self loops
}

// ---------------------------------------------------------------------------
// pass 1: segment-max of leaky-relu logits over dst
// ---------------------------------------------------------------------------
__global__ void edge_max_k(const int* __restrict__ ei, int E, int N,
                           const float* __restrict__ als,
                           const float* __restrict__ ald,
                           float* __restrict__ nmax) {
  int e = blockIdx.x * blockDim.x + threadIdx.x;
  if (e >= E + N) return;
  int s, d; edge_ends(ei, E, e, s, d);
  const float4* sp = (const float4*)(als + (size_t)s * 8);
  const float4* dp = (const float4*)(ald + (size_t)d * 8);
  float4 s0 = sp[0], s1 = sp[1], d0 = dp[0], d1 = dp[1];
  float t[8] = { s0.x + d0.x, s0.y + d0.y, s0.z + d0.z, s0.w + d0.w,
                 s1.x + d1.x, s1.y + d1.y, s1.z + d1.z, s1.w + d1.w };
  #pragma unroll
  for (int h = 0; h < 8; ++h)
    atomicMaxF(nmax + (size_t)d * 8 + h, lrelu02(t[h]));
}

// ---------------------------------------------------------------------------
// pass 2: segment-sum of exp(e - max[dst]) over dst
// ---------------------------------------------------------------------------
__global__ void edge_denom_k(const int* __restrict__ ei, int E, int N,
                             const float* __restrict__ als,
                             const float* __restrict__ ald,
                             const float* __restrict__ nmax,
                             float* __restrict__ nden) {
  int e = blockIdx.x * blockDim.x + threadIdx.x;
  if (e >= E + N) return;
  int s, d; edge_ends(ei, E, e, s, d);
  const float4* sp = (const float4*)(als + (size_t)s * 8);
  const float4* dp = (const float4*)(ald + (size_t)d * 8);
  const float4* mp = (const float4*)(nmax + (size_t)d * 8);
  float4 s0 = sp[0], s1 = sp[1], d0 = dp[0], d1 = dp[1];
  float4 m0 = mp[0], m1 = mp[1];
  float t[8] = { s0.x + d0.x, s0.y + d0.y, s0.z + d0.z, s0.w + d0.w,
                 s1.x + d1.x, s1.y + d1.y, s1.z + d1.z, s1.w + d1.w };
  float m[8] = { m0.x, m0.y, m0.z, m0.w, m1.x, m1.y, m1.z, m1.w };
  #pragma unroll
  for (int h = 0; h < 8; ++h)
    atomicAdd(nden + (size_t)d * 8 + h, __expf(lrelu02(t[h]) - m[h]));
}

// ---------------------------------------------------------------------------
// pass 3: out[dst] += alpha * h[src]   (alpha = exp(e-max)/denom)
// ---------------------------------------------------------------------------
template <int C>
__global__ void edge_aggr_k(const int* __restrict__ ei, int E, int N,
                            const float* __restrict__ als,
                            const float* __restrict__ ald,
                            const float* __restrict__ nmax,
                            const float* __restrict__ nden,
                            const float* __restrict__ Hf,
                            float* __restrict__ Out) {
  int e = blockIdx.x * blockDim.x + threadIdx.x;
  if (e >= E + N) return;
  int s, d; edge_ends(ei, E, e, s, d);
  const float4* sp = (const float4*)(als + (size_t)s * 8);
  const float4* dp = (const float4*)(ald + (size_t)d * 8);
  const float4* mp = (const float4*)(nmax + (size_t)d * 8);
  const float4* np = (const float4*)(nden + (size_t)d * 8);
  float4 s0 = sp[0], s1 = sp[1], d0 = dp[0], d1 = dp[1];
  float4 m0 = mp[0], m1 = mp[1], n0 = np[0], n1 = np[1];
  float t[8] = { s0.x + d0.x, s0.y + d0.y, s0.z + d0.z, s0.w + d0.w,
                 s1.x + d1.x, s1.y + d1.y, s1.z + d1.z, s1.w + d1.w };
  float m[8] = { m0.x, m0.y, m0.z, m0.w, m1.x, m1.y, m1.z, m1.w };
  float dn[8] = { n0.x, n0.y, n0.z, n0.w, n1.x, n1.y, n1.z, n1.w };
  const float* hs = Hf + (size_t)s * HEADS * C;
  float* od = Out + (size_t)d * HEADS * C;
  #pragma unroll
  for (int h = 0; h < 8; ++h) {
    float alpha = __expf(lrelu02(t[h]) - m[h]) / dn[h];
    #pragma unroll
    for (int c = 0; c < C; ++c)
      atomicAdd(od + h * C + c, alpha * hs[h * C + c]);
  }
}

template <int HC>   // HC is a power of two (64 / 128)
__global__ void bias_elu_k(float* __restrict__ X, const float* __restrict__ b,
                           size_t total) {
  size_t i = (size_t)blockIdx.x * blockDim.x + threadIdx.x;
  if (i >= total) return;
  float v = X[i] + b[(int)(i & (size_t)(HC - 1))];
  X[i] = v > 0.f ? v : (__expf(v) - 1.f);
}

// ---------------------------------------------------------------------------
// graph pooling: per node scatter into per-graph sum / max / count
// ---------------------------------------------------------------------------
__global__ void pool_node_k(const float* __restrict__ X,
                            const int* __restrict__ batch,
                            float* __restrict__ gsum, float* __restrict__ gmax,
                            float* __restrict__ gcnt) {
  int n = blockIdx.x;
  int g = batch[n];
  int j = threadIdx.x;                       // 128 features
  float v = X[(size_t)n * 128 + j];
  atomicAdd(gsum + (size_t)g * 128 + j, v);
  atomicMaxF(gmax + (size_t)g * 128 + j, v);
  if (j == 0) atomicAdd(gcnt + g, 1.f);
}

__global__ void pool_final_k(const float* __restrict__ gsum,
                             const float* __restrict__ gmax,
                             const float* __restrict__ gcnt,
                             const float* __restrict__ fcW,
                             const float* __restrict__ fcb,
                             float* __restrict__ out) {
  __shared__ float feat[256];
  __shared__ float logit[6];
  int g = blockIdx.x;
  float cnt = gcnt[g];
  int j = threadIdx.x;                       // 128 threads
  feat[j]       = gsum[(size_t)g * 128 + j] / fmaxf(cnt, 1.0f);
  float mv      = gmax[(size_t)g * 128 + j];
  feat[128 + j] = (cnt > 0.f) ? mv : 0.f;
  __syncthreads();
  if (j < 6) {
    float acc = fcb[j];
    for (int c = 0; c < 256; ++c) acc = fmaf(feat[c], fcW[c * 6 + j], acc);
    logit[j] = acc;
  }
  __syncthreads();
  if (j == 0) {
    float mx = logit[0];
    #pragma unroll
    for (int k = 1; k < 6; ++k) mx = fmaxf(mx, logit[k]);
    float s = 0.f;
    #pragma unroll
    for (int k = 0; k < 6; ++k) s += __expf(logit[k] - mx);
    float lse = mx + __logf(s);
    #pragma unroll
    for (int k = 0; k < 6; ++k) out[(size_t)g * 6 + k] = logit[k] - lse;
  }
}

// ---------------------------------------------------------------------------
static inline unsigned cdiv(size_t a, unsigned b) { return (unsigned)((a + b - 1) / b); }

extern "C" void kernel_launch(void* const* d_in, const int* in_sizes, int n_in,
                              void* d_out, int out_size, void* d_ws, size_t ws_size,
                              hipStream_t stream) {
  const float* x   = (const float*)d_in[0];
  const int* ei    = (const int*)d_in[1];
  const int* batch = (const int*)d_in[2];
  const int N = in_sizes[0] / 18;
  const int E = in_sizes[1] / 2;
  const int G = out_size / 6;
  const int EL = E + N;
  const float NEG_INF = -__builtin_huge_valf();

  // workspace layout (floats). bufA/bufB ping-pong: x dies after GEMM, so the
  // aggregation output reuses its buffer.
  float* bufA = (float*)d_ws;                      // layer out / next-layer x
  float* bufB = bufA + (size_t)N * 128;            // h = xW
  float* als  = bufB + (size_t)N * 128;            // [N,8]
  float* ald  = als  + (size_t)N * 8;
  float* nmax = ald  + (size_t)N * 8;
  float* nden = nmax + (size_t)N * 8;
  float* gsum = nden + (size_t)N * 8;              // [G,128]
  float* gmax = gsum + (size_t)G * 128;
  float* gcnt = gmax + (size_t)G * 128;            // [G]

  // one GAT layer; CIN = input features, C = per-head channels (COUT = 8*C)
  auto layer = [&](const float* xin, auto cin_tag, auto c_tag, int wi) {
    constexpr int CIN = decltype(cin_tag)::value;
    constexpr int C   = decltype(c_tag)::value;
    constexpr int HC  = HEADS * C;
    const float* Wl = (const float*)d_in[wi];
    const float* aS = (const float*)d_in[wi + 1];
    const float* aD = (const float*)d_in[wi + 2];
    const float* bl = (const float*)d_in[wi + 3];
    dim3 gg(cdiv((size_t)N, 32), HC / 16);
    gemm16x4_wmma<CIN, HC><<<gg, 64, 0, stream>>>(xin, Wl, bufB, N);
    attn_logits_k<C><<<cdiv((size_t)N * 8, 256), 256, 0, stream>>>(
        bufB, aS, aD, als, ald, N);
    fillk<<<cdiv((size_t)N * 8, 256), 256, 0, stream>>>(nmax, NEG_INF, (size_t)N * 8);
    fillk<<<cdiv((size_t)N * 8, 256), 256, 0, stream>>>(nden, 0.f, (size_t)N * 8);
    fillk<<<cdiv((size_t)N * HC, 256), 256, 0, stream>>>(bufA, 0.f, (size_t)N * HC);
    edge_max_k<<<cdiv((size_t)EL, 256), 256, 0, stream>>>(ei, E, N, als, ald, nmax);
    edge_denom_k<<<cdiv((size_t)EL, 256), 256, 0, stream>>>(ei, E, N, als, ald, nmax, nden);
    edge_aggr_k<C><<<cdiv((size_t)EL, 256), 256, 0, stream>>>(
        ei, E, N, als, ald, nmax, nden, bufB, bufA);
    bias_elu_k<HC><<<cdiv((size_t)N * HC, 256), 256, 0, stream>>>(
        bufA, bl, (size_t)N * HC);
  };

  layer(x,    std::integral_constant<int, 18>{},  std::integral_constant<int, 8>{},  3);
  layer(bufA, std::integral_constant<int, 64>{},  std::integral_constant<int, 16>{}, 7);
  layer(bufA, std::integral_constant<int, 128>{}, std::integral_constant<int, 16>{}, 11);
  layer(bufA, std::integral_constant<int, 128>{}, std::integral_constant<int, 16>{}, 15);

  fillk<<<cdiv((size_t)G * 128, 256), 256, 0, stream>>>(gsum, 0.f, (size_t)G * 128);
  fillk<<<cdiv((size_t)G * 128, 256), 256, 0, stream>>>(gmax, NEG_INF, (size_t)G * 128);
  fillk<<<cdiv((size_t)G, 256), 256, 0, stream>>>(gcnt, 0.f, (size_t)G);
  pool_node_k<<<N, 128, 0, stream>>>(bufA, batch, gsum, gmax, gcnt);
  pool_final_k<<<G, 128, 0, stream>>>(gsum, gmax, gcnt,
                                      (const float*)d_in[19], (const float*)d_in[20],
                                      (float*)d_out);
}